// GATLayer_4269197492331
// MI455X (gfx1250) — compile-verified
//
#include <hip/hip_runtime.h>
#include <hip/hip_bf16.h>

// ---------------- problem constants (match reference) ----------------
#define NN    50000          // nodes
#define FIN   256            // input features
#define NH    4              // heads
#define FOUT  64             // features per head
#define HF    (NH * FOUT)    // 256 = output width
#define ALPHA 0.2f

typedef __attribute__((ext_vector_type(2))) float v2f;
typedef __attribute__((ext_vector_type(8))) float v8f;

// ---------------------------------------------------------------------
// Kernel 0: init out=0, segment-max = -FLT_MAX, denom = 0
// ---------------------------------------------------------------------
__global__ __launch_bounds__(256) void gat_init(float* __restrict__ out,
                                                float* __restrict__ mbuf,
                                                float* __restrict__ denom) {
  size_t i = (size_t)blockIdx.x * blockDim.x + threadIdx.x;
  if (i < (size_t)NN * HF) out[i] = 0.0f;
  if (i < (size_t)NN * NH) {
    mbuf[i]  = -3.402823466e38f;   // -FLT_MAX
    denom[i] = 0.0f;
  }
}

// ---------------------------------------------------------------------
// Kernel 1: h = x @ W   via V_WMMA_F32_16X16X4_F32 (exact fp32)
// One wave per 16x16 output tile. 3125 M-tiles x 16 N-tiles = 50000 tiles.
// A (16x4 f32) lane layout: lane l holds row (l&15), K = kb, kb+1, kb=(l>>4)*2
// B (4x16 f32) lane layout: lane l holds col (l&15), K = kb, kb+1
// C/D: VGPR r -> row r + (l>>4)*8, col (l&15)
// ---------------------------------------------------------------------
__global__ __launch_bounds__(256) void gat_gemm(const float* __restrict__ x,
                                                const float* __restrict__ W,
                                                float* __restrict__ h) {
  int wave = blockIdx.x * (blockDim.x >> 5) + (threadIdx.x >> 5);
  if (wave >= (NN / 16) * 16) return;           // 50000 tiles exactly
  int tm   = wave >> 4;                          // 0..3124
  int tn   = wave & 15;                          // 0..15
  int lane = threadIdx.x & 31;

  int mrow = tm * 16 + (lane & 15);
  int ncol = tn * 16 + (lane & 15);
  int kb   = (lane >> 4) * 2;

  const float* __restrict__ xrow = x + (size_t)mrow * FIN + kb;  // 8B aligned
  const float* __restrict__ wcol = W + (size_t)kb * HF + ncol;

  v8f acc = {};
  #pragma unroll 4
  for (int k0 = 0; k0 < FIN; k0 += 4) {
    v2f a = *(const v2f*)(xrow + k0);            // x[mrow][k0+kb .. +1]
    v2f b;
    b.x = wcol[(size_t)k0 * HF];                 // W[k0+kb  ][ncol]
    b.y = wcol[(size_t)(k0 + 1) * HF];           // W[k0+kb+1][ncol]
    acc = __builtin_amdgcn_wmma_f32_16x16x4_f32(
        /*neg_a=*/false, a, /*neg_b=*/false, b,
        /*c_mod=*/(short)0, acc, /*reuse_a=*/false, /*reuse_b=*/false);
  }

  int rbase = tm * 16 + (lane >> 4) * 8;
  int c     = tn * 16 + (lane & 15);
  #pragma unroll
  for (int r = 0; r < 8; ++r)
    h[(size_t)(rbase + r) * HF + c] = acc[r];
}

// ---------------------------------------------------------------------
// Kernel 2: h_l[n][hd] = sum_f a_l[hd][f]*h[n][hd][f]  (same for h_r)
// One thread per (node, head).
// ---------------------------------------------------------------------
__global__ __launch_bounds__(256) void gat_proj(const float* __restrict__ h,
                                                const float* __restrict__ a_l,
                                                const float* __restrict__ a_r,
                                                float* __restrict__ hl,
                                                float* __restrict__ hr) {
  int t = blockIdx.x * blockDim.x + threadIdx.x;
  if (t >= NN * NH) return;
  int n  = t >> 2;
  int hd = t & 3;
  const float4* hp = (const float4*)(h + (size_t)n * HF + hd * FOUT);
  const float4* al = (const float4*)(a_l + hd * FOUT);
  const float4* ar = (const float4*)(a_r + hd * FOUT);
  float sl = 0.0f, sr = 0.0f;
  #pragma unroll
  for (int i = 0; i < FOUT / 4; ++i) {
    float4 hv = hp[i], av = al[i], bv = ar[i];
    sl += hv.x * av.x + hv.y * av.y + hv.z * av.z + hv.w * av.w;
    sr += hv.x * bv.x + hv.y * bv.y + hv.z * bv.z + hv.w * bv.w;
  }
  hl[t] = sl;
  hr[t] = sr;
}

// ---------------------------------------------------------------------
// Kernel 3: per-(edge,head) leaky-relu logit + atomic segment max by row.
// Float atomic-max via signed-int max (v>=0) / unsigned-int min (v<0).
// ---------------------------------------------------------------------
__global__ __launch_bounds__(256) void gat_edge_logits(const int* __restrict__ erow,
                                                       const int* __restrict__ ecol,
                                                       const float* __restrict__ hl,
                                                       const float* __restrict__ hr,
                                                       float* __restrict__ ebuf,
                                                       float* __restrict__ mbuf,
                                                       int E) {
  int t = blockIdx.x * blockDim.x + threadIdx.x;
  if (t >= E * NH) return;
  int e  = t >> 2;
  int hd = t & 3;
  int r  = erow[e];
  int c  = ecol[e];
  float s = hl[r * NH + hd] + hr[c * NH + hd];
  float v = s > 0.0f ? s : ALPHA * s;
  ebuf[t] = v;
  int* addr = (int*)(mbuf + r * NH + hd);
  if (v >= 0.0f) atomicMax(addr, __float_as_int(v));
  else           atomicMin((unsigned int*)addr, __float_as_uint(v));
}

// ---------------------------------------------------------------------
// Kernel 4: one wave per edge.
//   ex[hd] = exp(e - m[row]); denom[row][hd] += ex[hd];
//   out[row][hd][f] += ex[hd] * h[col][hd][f]   (unnormalized accumulate)
// Gather of h[col] and scatter atomics are lane-coalesced (128B/step).
// ---------------------------------------------------------------------
__global__ __launch_bounds__(256) void gat_edge_scatter(const int* __restrict__ erow,
                                                        const int* __restrict__ ecol,
                                                        const float* __restrict__ ebuf,
                                                        const float* __restrict__ mbuf,
                                                        const float* __restrict__ h,
                                                        float* __restrict__ denom,
                                                        float* __restrict__ out,
                                                        int E) {
  int wave = blockIdx.x * (blockDim.x >> 5) + (threadIdx.x >> 5);
  if (wave >= E) return;
  int lane = threadIdx.x & 31;
  int r = erow[wave];
  int c = ecol[wave];

  float ex[NH];
  #pragma unroll
  for (int hd = 0; hd < NH; ++hd) {
    float v = ebuf[(size_t)wave * NH + hd];
    float m = mbuf[r * NH + hd];
    ex[hd] = __expf(v - m);
  }
  if (lane < NH) atomicAdd(&denom[r * NH + lane], ex[lane]);

  const float* __restrict__ hc = h + (size_t)c * HF;
  float* __restrict__ orow     = out + (size_t)r * HF;
  #pragma unroll
  for (int j = lane; j < HF; j += 32) {
    int hd = j >> 6;
    atomicAdd(&orow[j], ex[hd] * hc[j]);
  }
}

// ---------------------------------------------------------------------
// Kernel 5: out[n][hd][f] /= max(denom[n][hd], 1e-16)
// ---------------------------------------------------------------------
__global__ __launch_bounds__(256) void gat_norm(float* __restrict__ out,
                                                const float* __restrict__ denom) {
  size_t t = (size_t)blockIdx.x * blockDim.x + threadIdx.x;
  if (t >= (size_t)NN * HF) return;
  int n  = (int)(t >> 8);          // HF == 256
  int hd = (int)((t >> 6) & 3);    // FOUT == 64
  float d = denom[n * NH + hd];
  out[t] = out[t] / fmaxf(d, 1e-16f);
}

// ---------------------------------------------------------------------
extern "C" void kernel_launch(void* const* d_in, const int* in_sizes, int n_in,
                              void* d_out, int out_size, void* d_ws, size_t ws_size,
                              hipStream_t stream) {
  (void)n_in; (void)out_size; (void)ws_size;

  const float* x   = (const float*)d_in[0];     // [N, 256]
  const int*   edg = (const int*)d_in[1];       // [2, E]
  const float* W   = (const float*)d_in[2];     // [256, 256]
  const float* a_l = (const float*)d_in[3];     // [1, 4, 64]
  const float* a_r = (const float*)d_in[4];     // [1, 4, 64]
  float*       out = (float*)d_out;             // [N, 256]

  const int E = in_sizes[1] / 2;
  const int* erow = edg;
  const int* ecol = edg + E;

  // workspace layout (floats)
  float* h     = (float*)d_ws;                   // N*HF   = 12.8M floats
  float* hl    = h     + (size_t)NN * HF;        // N*NH
  float* hr    = hl    + (size_t)NN * NH;
  float* mbuf  = hr    + (size_t)NN * NH;
  float* denom = mbuf  + (size_t)NN * NH;
  float* ebuf  = denom + (size_t)NN * NH;        // E*NH

  // 0) init (out zero, max=-inf, denom zero)
  {
    int blocks = (NN * HF + 255) / 256;
    gat_init<<<blocks, 256, 0, stream>>>(out, mbuf, denom);
  }
  // 1) h = x @ W  (WMMA f32)
  {
    int tiles  = (NN / 16) * 16;                 // 50000 waves
    int blocks = (tiles + 7) / 8;                // 8 waves / block
    gat_gemm<<<blocks, 256, 0, stream>>>(x, W, h);
  }
  // 2) attention projections
  {
    int blocks = (NN * NH + 255) / 256;
    gat_proj<<<blocks, 256, 0, stream>>>(h, a_l, a_r, hl, hr);
  }
  // 3) edge logits + segment max
  {
    int blocks = (E * NH + 255) / 256;
    gat_edge_logits<<<blocks, 256, 0, stream>>>(erow, ecol, hl, hr, ebuf, mbuf, E);
  }
  // 4) exp / denom / unnormalized scatter-accumulate
  {
    int blocks = (E + 7) / 8;                    // 1 wave per edge
    gat_edge_scatter<<<blocks, 256, 0, stream>>>(erow, ecol, ebuf, mbuf, h,
                                                 denom, out, E);
  }
  // 5) normalize
  {
    int blocks = (NN * HF + 255) / 256;
    gat_norm<<<blocks, 256, 0, stream>>>(out, denom);
  }
}